// VQ_28707561406858
// MI455X (gfx1250) — compile-verified
//
#include <hip/hip_runtime.h>

// CDNA5 / gfx1250 VQ-VAE vector quantization.
// out[0:8388608] = quantised (straight-through output == quantised numerically)
// out[8388608]   = 1.25 * mean((q - x)^2)

typedef __attribute__((ext_vector_type(2))) float v2f;
typedef __attribute__((ext_vector_type(8))) float v8f;

#define ROWS     131072      // 32*64*64*64 / 64
#define DIM      64          // latent dim (GEMM K)
#define CODES    512         // codebook entries (GEMM N)
#define CB_PITCH 520         // padded LDS row pitch (floats); 520%64==8 -> no bank conflicts
#define NBLOCK   1024
#define WAVES    8           // 256 threads = 8 wave32

__global__ __launch_bounds__(256) void vq_main(const float* __restrict__ x,
                                               const float* __restrict__ cb,
                                               float* __restrict__ out,
                                               float* __restrict__ bsum)
{
    __shared__ float lcb[DIM * CB_PITCH];   // padded codebook copy (130 KB)
    __shared__ float nh[CODES];             // 0.5 * ||e_k||^2
    __shared__ int   idxbuf[WAVES][16];     // winning code per row, per wave
    __shared__ float wsum[WAVES];           // per-wave squared-error partials

    const int t    = threadIdx.x;
    const int wave = t >> 5;
    const int lane = t & 31;
    const int lo   = lane & 15;             // M (A/C) or N (B/C) position
    const int hi2  = (lane >> 4) * 2;       // K group offset for A/B fragments

    // ---- stage codebook (64 x 512 f32, row-major) into padded LDS via float4 ----
    {
        const float4* src = (const float4*)cb;
        float4*       dst = (float4*)lcb;
        #pragma unroll
        for (int it = 0; it < 32; ++it) {
            int q4 = it * 256 + t;          // 8192 float4 total
            int d  = q4 >> 7;               // row   (q4 / 128)
            int c4 = q4 & 127;              // col/4
            dst[d * (CB_PITCH / 4) + c4] = src[q4];
        }
    }
    __syncthreads();

    // ---- precompute 0.5*||e_k||^2 ----
    for (int k = t; k < CODES; k += 256) {
        float s = 0.f;
        #pragma unroll
        for (int d = 0; d < DIM; ++d) { float e = lcb[d * CB_PITCH + k]; s += e * e; }
        nh[k] = 0.5f * s;
    }
    __syncthreads();

    // ---- each wave handles one 16-row slab ----
    const int slab = blockIdx.x * WAVES + wave;   // 8192 slabs exactly
    const int r0   = slab * 16;

    // A fragments: 16x4 f32 tile per kt.
    // ISA layout: lane holds (M = lane%16, K = kt*4 + (lane/16)*2 + j) in VGPR j.
    v2f a[16];
    #pragma unroll
    for (int kt = 0; kt < 16; ++kt) {
        a[kt] = *(const v2f*)(x + (r0 + lo) * DIM + kt * 4 + hi2);
    }

    float bestv[8];
    int   besti[8];
    #pragma unroll
    for (int v = 0; v < 8; ++v) { bestv[v] = -3.4e38f; besti[v] = 0; }

    // score s_k = x.e_k - 0.5||e_k||^2 ; argmin dist == argmax s
    for (int nt = 0; nt < 32; ++nt) {
        const int n0 = nt * 16;
        const float bias = -nh[n0 + lo];
        v8f c;
        #pragma unroll
        for (int v = 0; v < 8; ++v) c[v] = bias;

        #pragma unroll
        for (int kt = 0; kt < 16; ++kt) {
            // B fragment 4x16: lane holds (K = kt*4 + hi2 + j, N = n0 + lane%16) in VGPR j
            const float* bp = &lcb[(kt * 4 + hi2) * CB_PITCH + n0 + lo];
            v2f b;
            b[0] = bp[0];
            b[1] = bp[CB_PITCH];
            c = __builtin_amdgcn_wmma_f32_16x16x4_f32(
                    false, a[kt], false, b, (short)0, c, false, false);
        }

        // per-lane running argmax; candidate indices increase with nt, so
        // strict '>' preserves the first-(lowest-index)-wins tie rule per lane.
        const int n = n0 + lo;
        #pragma unroll
        for (int v = 0; v < 8; ++v) {
            if (c[v] > bestv[v]) { bestv[v] = c[v]; besti[v] = n; }
        }
    }

    // ---- cross-lane argmax within each 16-lane N-group (xor masks stay in-group) ----
    #pragma unroll
    for (int v = 0; v < 8; ++v) {
        float bv = bestv[v];
        int   bi = besti[v];
        #pragma unroll
        for (int m = 1; m <= 8; m <<= 1) {
            float ov = __shfl_xor(bv, m, 32);
            int   oi = __shfl_xor(bi, m, 32);
            if (ov > bv || (ov == bv && oi < bi)) { bv = ov; bi = oi; }
        }
        besti[v] = bi;
    }

    // lane 0 holds winners for rows M=0..7 (VGPR v), lane 16 for rows M=8..15
    if (lane == 0) {
        #pragma unroll
        for (int v = 0; v < 8; ++v) idxbuf[wave][v] = besti[v];
    }
    if (lane == 16) {
        #pragma unroll
        for (int v = 0; v < 8; ++v) idxbuf[wave][8 + v] = besti[v];
    }
    asm volatile("s_wait_dscnt 0" ::: "memory");   // wave-internal LDS visibility

    // ---- gather quantised rows, write output, accumulate squared error ----
    const int  r    = lo;
    const int  half = lane >> 4;                   // 2 lanes per row: d halves
    const int  kq   = idxbuf[wave][r];
    const int  row  = r0 + r;
    float acc = 0.f;
    #pragma unroll
    for (int i = 0; i < 32; ++i) {
        int   d  = half * 32 + i;
        float q  = lcb[d * CB_PITCH + kq];
        float xi = x[row * DIM + d];
        out[row * DIM + d] = q;
        float e = q - xi;
        acc += e * e;
    }

    // deterministic loss reduction: wave butterfly -> fixed-order block sum
    #pragma unroll
    for (int m = 1; m <= 16; m <<= 1) acc += __shfl_xor(acc, m, 32);
    if (lane == 0) wsum[wave] = acc;
    __syncthreads();
    if (t == 0) {
        float s = 0.f;
        #pragma unroll
        for (int w = 0; w < WAVES; ++w) s += wsum[w];
        bsum[blockIdx.x] = s;
    }
}

__global__ __launch_bounds__(256) void vq_loss(const float* __restrict__ bsum,
                                               float* __restrict__ out)
{
    __shared__ float part[256];
    const int t = threadIdx.x;
    float s = 0.f;
    #pragma unroll
    for (int j = 0; j < NBLOCK / 256; ++j) s += bsum[j * 256 + t];  // fixed order
    part[t] = s;
    __syncthreads();
    if (t == 0) {
        float tot = 0.f;
        for (int i = 0; i < 256; ++i) tot += part[i];               // fixed order
        out[ROWS * DIM] = 1.25f * tot * (1.0f / (float)(ROWS * DIM));
    }
}

extern "C" void kernel_launch(void* const* d_in, const int* in_sizes, int n_in,
                              void* d_out, int out_size, void* d_ws, size_t ws_size,
                              hipStream_t stream)
{
    (void)in_sizes; (void)n_in; (void)out_size; (void)ws_size;
    const float* x  = (const float*)d_in[0];   // inputs  (32,64,64,64) f32
    const float* cb = (const float*)d_in[1];   // encoded (64,512) f32
    float* out = (float*)d_out;                // [8388608 quantised] + [1 loss]
    float* ws  = (float*)d_ws;                 // >= NBLOCK floats of scratch

    vq_main<<<NBLOCK, 256, 0, stream>>>(x, cb, out, ws);
    vq_loss<<<1, 256, 0, stream>>>(ws, out);
}